// ChunkedQuantHead_10788957847687
// MI455X (gfx1250) — compile-verified
//
#include <hip/hip_runtime.h>
#include <hip/hip_bf16.h>
#include <stdint.h>

// ---------------------------------------------------------------------------
// ChunkedQuantHead on MI455X (gfx1250), single-pass-over-x design.
//
// Roofline: x is 16384x4096 f32 = 268 MB; total useful FLOPs < 0.4 GFLOP ->
// pure HBM bound (23.3 TB/s => ~11.6 us floor). We stream x exactly once:
//   - Tensor DMA path: each wave async-copies its 16x256 f32 chunk tile into
//     LDS (global_load_async_to_lds_b128, ASYNCcnt, double buffered, no
//     barriers since buffers are wave-private), then
//   - computes ALL 16 per-chunk partial GEMMs with V_WMMA_F32_16X16X4_F32
//     (exact f32 math, free relative to bandwidth) from conflict-free
//     ds_load_b64 fragments, plus the per-chunk mean-|x| reduction.
// Data-dependent top-k / 1-bit weight selection runs in a tiny follow-up
// kernel; a cheap combine kernel applies the 10x10 head.
// ---------------------------------------------------------------------------

typedef __attribute__((ext_vector_type(2))) float v2f;
typedef __attribute__((ext_vector_type(8))) float v8f;

#define IN_FEATS 4096
#define CHUNKS   16
#define TOPK     4
#define OUT_N    10
#define CHUNK    256
#define BATCH    16384

// LDS staging: 16 rows x 260 floats (pitch padded 256->260 so the WMMA
// fragment read lmod*260 + k hits all 64 banks exactly once -> conflict-free)
#define ROWPITCH_F 260
#define ROWPITCH_B (ROWPITCH_F * 4)        // 1040 B
#define BUF_BYTES  (16 * ROWPITCH_B)       // 16640 B per buffer
#define WAVE_LDS   (2 * BUF_BYTES)         // double buffered per wave
#define WAVES_PER_BLOCK 4
#define BLOCK_LDS  (WAVES_PER_BLOCK * WAVE_LDS)   // 133120 B (< 320 KB/WG)

// ws layout (bytes)
#define WS_ACTS    0        // 16 floats: sum of |x| per chunk
#define WS_TOPIDX  64       // 4 ints
#define WS_WEFF    128      // 100 floats (10x10 effective head weights)
#define WS_WPAD    1024     // padded expert weights [16][16][256] f32 (256 KB)
#define WS_ROUTED  264192   // BATCH*16*10 floats: per-chunk partial GEMMs

// ---------------------------------------------------------------------------
// Kernel 0: zero the 16 act accumulators (fresh every launch; ws is not
// re-poisoned between replays, so we must not rely on prior contents).
// ---------------------------------------------------------------------------
__global__ void cqh_zero_acts(float* acts_sum) {
    if (threadIdx.x < CHUNKS) acts_sum[threadIdx.x] = 0.0f;
}

// ---------------------------------------------------------------------------
// Kernel 0b: build zero-padded B operand wpad[c][n][k]:
//   n in [0,16): rows 10..15 are zero -> WMMA B loads are branch-free
//   (kills the per-iteration s_and_saveexec/exec restore seen in round 1).
// ---------------------------------------------------------------------------
__global__ void __launch_bounds__(256)
cqh_pad_weights(const float* __restrict__ expert_w, float* __restrict__ wpad) {
    int idx = blockIdx.x * blockDim.x + threadIdx.x;   // 16*16*256 = 65536
    if (idx >= CHUNKS * 16 * CHUNK) return;
    int k = idx & (CHUNK - 1);
    int n = (idx >> 8) & 15;
    int c = idx >> 12;
    float v = 0.0f;
    if (n < OUT_N) v = expert_w[((size_t)c * OUT_N + n) * CHUNK + k];
    wpad[idx] = v;
}

// ---------------------------------------------------------------------------
// Async DMA of one 16x256 f32 chunk tile into a wave-private LDS buffer.
// 32 x global_load_async_to_lds_b128 per lane-wave group: q = j*32+lane,
// row r = q/64, float4-col k4 = q%64  -> 512 B fully-coalesced per instr.
// ASYNCcnt += 32 per call (6-bit counter, max 63: fits with double buffer).
// ---------------------------------------------------------------------------
__device__ __forceinline__ void cqh_async_fill(const float* __restrict__ xbase,
                                               uint32_t lds_base, int lane) {
    for (int j = 0; j < 32; ++j) {
        int q  = j * 32 + lane;
        int r  = q >> 6;
        int k4 = q & 63;
        const float* g = xbase + (size_t)r * IN_FEATS + (size_t)k4 * 4;
        uint32_t l = lds_base + (uint32_t)(r * ROWPITCH_B + k4 * 16);
        asm volatile("global_load_async_to_lds_b128 %0, %1, off"
                     :: "v"(l), "v"(g) : "memory");
    }
}

// ---------------------------------------------------------------------------
// Kernel 1: fused streaming pass over x.
//   - one wave (32 lanes) owns a tile of 16 batch rows (1024 tiles total)
//   - chunk c+1 is DMA'd to LDS while chunk c runs 64 WMMAs
//   - |x| accumulated per lane, wave32-reduced, one f32 atomic per chunk
//
// WMMA f32 16x16x4 operand layout (ISA 7.12.2, 32-bit A 16x4):
//   lane l (l<16):  A vgpr0 = x[row=l, K=k0+0], vgpr1 = K=k0+1
//   lane l (l>=16): A vgpr0 = x[row=l-16, K=k0+2], vgpr1 = K=k0+3
// => each lane reads a float2 at column k0 + 2*(l/16).  B (4x16) mirrors this
// with N = l%16.  C/D: vgpr r = row r + 8*(l/16), col l%16.
// ---------------------------------------------------------------------------
__global__ void __launch_bounds__(128)
cqh_fused_pass(const float* __restrict__ x,
               const float* __restrict__ wpad,
               float* __restrict__ acts_sum,
               float* __restrict__ routed_all) {
    extern __shared__ char smem[];

    const int lane = threadIdx.x & 31;
    const int wave = threadIdx.x >> 5;
    const int tile = blockIdx.x * WAVES_PER_BLOCK + wave;  // 16-row batch tile
    const int row0 = tile * 16;
    const int half = lane >> 4;                 // 0: lanes 0-15, 1: lanes 16-31
    const int lmod = lane & 15;

    const char* wbase = smem + wave * WAVE_LDS;            // wave-private LDS
    const uint32_t wbase_lds = (uint32_t)(uintptr_t)wbase; // low 32 = LDS addr

    const float* xtile = x + (size_t)row0 * IN_FEATS;      // tile row block

    // prime the pipeline: chunk 0 -> buffer 0
    cqh_async_fill(xtile, wbase_lds, lane);

    for (int c = 0; c < CHUNKS; ++c) {
        // WAR guard: all ds reads of the buffer we are about to overwrite
        // were consumed last iteration; make it explicit, then DMA c+1 and
        // wait for chunk c's 32 async ops (<= 32 outstanding = only c+1).
        if (c + 1 < CHUNKS) {
            asm volatile("s_wait_dscnt 0x0" ::: "memory");
            cqh_async_fill(xtile + (size_t)(c + 1) * CHUNK,
                           wbase_lds + (uint32_t)(((c + 1) & 1) * BUF_BYTES),
                           lane);
            asm volatile("s_wait_asynccnt 0x20" ::: "memory");
        } else {
            asm volatile("s_wait_asynccnt 0x0" ::: "memory");
        }

        // A fragments from LDS (conflict-free: pitch 260 floats)
        const char* arow = wbase + (c & 1) * BUF_BYTES
                         + lmod * ROWPITCH_B + (2 * half) * 4;
        // B fragments from padded weights (branch-free, L0/L2 resident)
        const float* brow = wpad + ((size_t)c * 16 + lmod) * CHUNK + 2 * half;

        v8f acc = {};          // routed partial (16 rows x 16 cols)
        float asum = 0.0f;     // per-lane sum of |x| for this chunk

#pragma unroll 8
        for (int k0 = 0; k0 < CHUNK; k0 += 4) {
            v2f a = *(const v2f*)(arow + (size_t)k0 * 4);   // ds_load_b64
            v2f b = *(const v2f*)(brow + k0);               // global_load_b64
            asum += __builtin_fabsf(a.x) + __builtin_fabsf(a.y);
            // D = A(16x4) * B(4x16) + C   -> v_wmma_f32_16x16x4_f32
            acc = __builtin_amdgcn_wmma_f32_16x16x4_f32(
                /*neg_a=*/false, a, /*neg_b=*/false, b,
                /*c_mod=*/(short)0, acc, /*reuse_a=*/false, /*reuse_b=*/false);
        }

        // wave32 tree reduction of |x| sums, one f32 atomic per wave per chunk
        for (int off = 16; off >= 1; off >>= 1)
            asum += __shfl_xor(asum, off, 32);
        if (lane == 0) atomicAdd(&acts_sum[c], asum);

        // scatter accumulator: vgpr r holds row (r + 8*half), col lmod
        if (lmod < OUT_N) {
            float* dst = routed_all +
                ((size_t)(row0 + 8 * half) * CHUNKS + c) * OUT_N + lmod;
#pragma unroll
            for (int r = 0; r < 8; ++r)
                dst[(size_t)r * CHUNKS * OUT_N] = acc[r];
        }
    }
}

// ---------------------------------------------------------------------------
// Kernel 2: finalize routing decision (tiny, one lane).
//   acts = acts_sum / (BATCH*CHUNK); top-4 (desc, ties -> lowest index, same
//   as jax.lax.top_k); scalar_act = max(acts); build w_eff per THRESH branch.
// ---------------------------------------------------------------------------
__global__ void cqh_finalize(const float* __restrict__ acts_sum,
                             const float* __restrict__ quant_w,
                             int* __restrict__ top_idx,
                             float* __restrict__ w_eff) {
    if (threadIdx.x != 0) return;
    float acts[CHUNKS];
    float scalar_act = -3.0e38f;
    for (int i = 0; i < CHUNKS; ++i) {
        acts[i] = acts_sum[i] * (1.0f / ((float)BATCH * (float)CHUNK));
        if (acts[i] > scalar_act) scalar_act = acts[i];
    }
    bool used[CHUNKS] = {};
    for (int t = 0; t < TOPK; ++t) {
        int best = 0; float bv = -3.0e38f;
        for (int i = 0; i < CHUNKS; ++i)
            if (!used[i] && acts[i] > bv) { bv = acts[i]; best = i; }
        used[best] = true;
        top_idx[t] = best;
    }
    float s = 0.0f;
    for (int i = 0; i < OUT_N * OUT_N; ++i) s += __builtin_fabsf(quant_w[i]);
    const float wmean = s * (1.0f / (OUT_N * OUT_N));
    const bool full = (scalar_act > 0.5f);
    for (int i = 0; i < OUT_N * OUT_N; ++i) {
        float w = quant_w[i];
        float sg = (w > 0.0f) ? 1.0f : ((w < 0.0f) ? -1.0f : 0.0f);
        w_eff[i] = full ? w : sg * wmean;
    }
}

// ---------------------------------------------------------------------------
// Kernel 3: combine. One thread per batch row: sum the 4 selected per-chunk
// partials (+expert_b), then apply the 10x10 effective head (+quant_b).
// ---------------------------------------------------------------------------
__global__ void __launch_bounds__(256)
cqh_combine(const float* __restrict__ routed_all,
            const int* __restrict__ top_idx,
            const float* __restrict__ w_eff,
            const float* __restrict__ expert_b,
            const float* __restrict__ quant_b,
            float* __restrict__ out) {
    const int b = blockIdx.x * blockDim.x + threadIdx.x;
    if (b >= BATCH) return;
    const int i0 = top_idx[0], i1 = top_idx[1], i2 = top_idx[2], i3 = top_idx[3];
    const float* rb = routed_all + (size_t)b * CHUNKS * OUT_N;
    float r[OUT_N];
#pragma unroll
    for (int o = 0; o < OUT_N; ++o)
        r[o] = rb[i0 * OUT_N + o] + rb[i1 * OUT_N + o] +
               rb[i2 * OUT_N + o] + rb[i3 * OUT_N + o] + expert_b[o];
#pragma unroll
    for (int oo = 0; oo < OUT_N; ++oo) {
        float s = quant_b[oo];
        const float* wrow = w_eff + oo * OUT_N;
#pragma unroll
        for (int o = 0; o < OUT_N; ++o) s = __builtin_fmaf(wrow[o], r[o], s);
        out[(size_t)b * OUT_N + oo] = s;
    }
}

// ---------------------------------------------------------------------------
extern "C" void kernel_launch(void* const* d_in, const int* in_sizes, int n_in,
                              void* d_out, int out_size, void* d_ws, size_t ws_size,
                              hipStream_t stream) {
    const float* x        = (const float*)d_in[0];   // [16384, 4096]
    const float* expert_w = (const float*)d_in[1];   // [16, 10, 256]
    const float* expert_b = (const float*)d_in[2];   // [10]
    const float* quant_w  = (const float*)d_in[3];   // [10, 10]
    const float* quant_b  = (const float*)d_in[4];   // [10]
    float* out = (float*)d_out;                      // [16384, 10]

    char* ws = (char*)d_ws;
    float* acts_sum   = (float*)(ws + WS_ACTS);
    int*   top_idx    = (int*)  (ws + WS_TOPIDX);
    float* w_eff      = (float*)(ws + WS_WEFF);
    float* wpad       = (float*)(ws + WS_WPAD);      // [16][16][256] f32
    float* routed_all = (float*)(ws + WS_ROUTED);    // BATCH*16*10 f32

    cqh_zero_acts<<<1, 32, 0, stream>>>(acts_sum);

    cqh_pad_weights<<<(CHUNKS * 16 * CHUNK + 255) / 256, 256, 0, stream>>>(
        expert_w, wpad);

    // 1024 tiles of 16 rows, 4 waves (tiles) per 128-thread block,
    // 130 KB dynamic LDS per workgroup for double-buffered async staging.
    cqh_fused_pass<<<BATCH / (16 * WAVES_PER_BLOCK), 128, BLOCK_LDS, stream>>>(
        x, wpad, acts_sum, routed_all);

    cqh_finalize<<<1, 32, 0, stream>>>(acts_sum, quant_w, top_idx, w_eff);

    cqh_combine<<<(BATCH + 255) / 256, 256, 0, stream>>>(
        routed_all, top_idx, w_eff, expert_b, quant_b, out);
}